// GuidedGATRegression_70145405878346
// MI455X (gfx1250) — compile-verified
//
#include <hip/hip_runtime.h>
#include <hip/hip_bf16.h>
#include <math.h>

// ---------------------------------------------------------------------------
// GuidedGATRegression on MI455X (gfx1250, wave32, WMMA)
//
//   L1: rw/log-reduce -> GEMM (v_wmma_f32_16x16x4_f32, 16x64 strip/wave) ->
//       node attn halves -> edge alpha+atomicMax -> edge exp+atomicSum ->
//       float4 scatter-aggregate (f32 atomics into L2-resident 51MB) -> ELU
//   L2: same with H=1 -> fused ELU + fc dot (wave32-per-node reduce)
// ---------------------------------------------------------------------------

#define NN 50000
#define EE 800000
#define DIN 128
#define HID 64
#define H1 4
#define F1 (H1 * HID)   // 256
#define NEG_SLOPE 0.01f
#define FGW 0.1f
#define ORD_NEG_INF 0x007FFFFFu  // f2ord(-inf)

typedef float v2f __attribute__((ext_vector_type(2)));
typedef float v4f __attribute__((ext_vector_type(4)));
typedef float v8f __attribute__((ext_vector_type(8)));

__device__ __forceinline__ unsigned f2ord(float f) {
  unsigned u = __float_as_uint(f);
  return (u & 0x80000000u) ? ~u : (u | 0x80000000u);
}
__device__ __forceinline__ float ord2f(unsigned u) {
  return (u & 0x80000000u) ? __uint_as_float(u ^ 0x80000000u) : __uint_as_float(~u);
}
__device__ __forceinline__ float eluf(float v) { return v > 0.f ? v : (expf(v) - 1.f); }

// -------------------------------- fills ------------------------------------
__global__ void fill_f32(float* p, float v, int n) {
  int i = blockIdx.x * blockDim.x + threadIdx.x;
  if (i < n) p[i] = v;
}
__global__ void fill_u32(unsigned* p, unsigned v, int n) {
  int i = blockIdx.x * blockDim.x + threadIdx.x;
  if (i < n) p[i] = v;
}

// --------------------- reward weight + deterministic sum -------------------
__global__ void rw_reduce(const float* __restrict__ X, int N, int F,
                          float* __restrict__ rw, float* __restrict__ sumOut) {
  __shared__ float sm[256];
  float acc = 0.f;
  for (int n = threadIdx.x; n < N; n += blockDim.x) {
    float v = 1.0f / (fabsf(X[(size_t)n * F + (F - 1)]) + 1e-6f);
    rw[n] = v;
    acc += v;
  }
  sm[threadIdx.x] = acc;
  __syncthreads();
  for (int s = blockDim.x >> 1; s > 0; s >>= 1) {
    if (threadIdx.x < s) sm[threadIdx.x] += sm[threadIdx.x + s];
    __syncthreads();
  }
  if (threadIdx.x == 0) sumOut[0] = sm[0];
}

__global__ void log_rw(const float* __restrict__ rw, const float* __restrict__ sum,
                       float* __restrict__ lg, int N) {
  int n = blockIdx.x * blockDim.x + threadIdx.x;
  if (n < N) lg[n] = logf(rw[n]) - logf(fmaxf(sum[0], 1e-12f));
}

// ------------------------- WMMA fp32 GEMM ----------------------------------
// C[M x F] = A[M x K] * B[K x F].  M%16==0, K%4==0, F%64==0.
// One wave computes a 16x64 strip (4 n-tiles reuse each A fragment).
// All 9 loads of a k-step are issued before the 4 WMMAs so the compiler can
// clause them and stagger loadcnt waits; addresses advance incrementally.
__global__ void gemm_wmma_f32(const float* __restrict__ A, const float* __restrict__ B,
                              float* __restrict__ C, int M, int K, int F) {
  const int lane = threadIdx.x & 31;
  const int wave = (blockIdx.x * blockDim.x + threadIdx.x) >> 5;
  const int nq = F >> 6;
  const int mtiles = M >> 4;
  if (wave >= mtiles * nq) return;                 // wave-uniform
  const int mt = wave / nq;
  const int q = wave - mt * nq;
  const int row = mt * 16 + (lane & 15);
  const int khalf = (lane >> 4) * 2;               // lanes 0-15 -> K 0,1 ; 16-31 -> K 2,3
  const int colBase = q * 64 + (lane & 15);

  const float* Ap = A + (size_t)row * K + khalf;
  const float* Bp = B + (size_t)khalf * F + colBase;

  v8f acc[4] = {v8f{}, v8f{}, v8f{}, v8f{}};
  const int ksteps = K >> 2;
#pragma unroll 2
  for (int ks = 0; ks < ksteps; ++ks) {
    v2f a, b0, b1, b2, b3;
    a.x = Ap[0];  a.y = Ap[1];
    b0.x = Bp[0];        b0.y = Bp[F];
    b1.x = Bp[16];       b1.y = Bp[F + 16];
    b2.x = Bp[32];       b2.y = Bp[F + 32];
    b3.x = Bp[48];       b3.y = Bp[F + 48];
    Ap += 4;
    Bp += (size_t)4 * F;
    acc[0] = __builtin_amdgcn_wmma_f32_16x16x4_f32(false, a, false, b0, (short)0, acc[0], false, false);
    acc[1] = __builtin_amdgcn_wmma_f32_16x16x4_f32(false, a, false, b1, (short)0, acc[1], false, false);
    acc[2] = __builtin_amdgcn_wmma_f32_16x16x4_f32(false, a, false, b2, (short)0, acc[2], false, false);
    acc[3] = __builtin_amdgcn_wmma_f32_16x16x4_f32(false, a, false, b3, (short)0, acc[3], false, false);
  }
  // C/D layout: VGPR r -> row r (lanes 0-15) / row 8+r (lanes 16-31), col = lane&15
  const int rbase = mt * 16 + (lane >> 4) * 8;
#pragma unroll
  for (int t = 0; t < 4; ++t) {
    const int col = q * 64 + t * 16 + (lane & 15);
#pragma unroll
    for (int r = 0; r < 8; ++r) {
      C[(size_t)(rbase + r) * F + col] = acc[t][r];
    }
  }
}

// ----------------- per-node attention halves al/ar (float4) ----------------
__global__ void node_attn(const float* __restrict__ h, const float* __restrict__ attL,
                          const float* __restrict__ attR, float* __restrict__ al,
                          float* __restrict__ ar, int N, int H, int C) {
  int t = blockIdx.x * blockDim.x + threadIdx.x;
  if (t >= N * H) return;
  int n = t / H, hh = t - n * H;
  const v4f* hp = (const v4f*)(h + (size_t)n * H * C + (size_t)hh * C);
  const v4f* lp = (const v4f*)(attL + hh * C);
  const v4f* rp = (const v4f*)(attR + hh * C);
  float sl = 0.f, sr = 0.f;
  for (int c = 0; c < (C >> 2); ++c) {
    v4f v = hp[c], l = lp[c], r = rp[c];
    sl += v.x * l.x + v.y * l.y + v.z * l.z + v.w * l.w;
    sr += v.x * r.x + v.y * r.y + v.z * r.z + v.w * r.w;
  }
  al[t] = sl;
  ar[t] = sr;
}

// ------------- edge pass A: raw alpha + segment atomic max -----------------
__global__ void edge_alpha(const int* __restrict__ src, const int* __restrict__ dst,
                           const float* __restrict__ al, const float* __restrict__ ar,
                           const float* __restrict__ lg, float* __restrict__ raw,
                           unsigned* __restrict__ mOrd, int E, int H) {
  int t = blockIdx.x * blockDim.x + threadIdx.x;
  if (t >= E * H) return;
  int e = t / H, hh = t - e * H;
  int s = src[e], d = dst[e];
  float a = al[d * H + hh] + ar[s * H + hh];
  a = a > 0.f ? a : NEG_SLOPE * a;         // leaky_relu
  a += FGW * lg[s];                        // reward guidance (src node)
  raw[t] = a;
  atomicMax(&mOrd[d * H + hh], f2ord(a));
}

// ------------- edge pass B: exp(alpha - max) + segment atomic sum ----------
__global__ void edge_exp(const int* __restrict__ dst, float* __restrict__ raw,
                         const unsigned* __restrict__ mOrd, float* __restrict__ sBuf,
                         int E, int H) {
  int t = blockIdx.x * blockDim.x + threadIdx.x;
  if (t >= E * H) return;
  int e = t / H, hh = t - e * H;
  int d = dst[e];
  float ev = expf(raw[t] - ord2f(mOrd[d * H + hh]));
  raw[t] = ev;
  atomicAdd(&sBuf[d * H + hh], ev);
}

// ------- edge pass C: float4 weighted scatter-aggregate (L2-resident) ------
// One thread = 4 consecutive channels (C%4==0, so a quad never crosses heads).
__global__ void edge_aggregate4(const int* __restrict__ src, const int* __restrict__ dst,
                                const float* __restrict__ h, const float* __restrict__ eBuf,
                                const float* __restrict__ sBuf, float* __restrict__ out,
                                int E, int H, int C) {
  const int Fq = (H * C) >> 2;                     // float4 quads per edge
  int t = blockIdx.x * blockDim.x + threadIdx.x;   // E*Fq <= 51.2M
  if (t >= E * Fq) return;
  int e = t / Fq, cq = t - e * Fq;
  int c = cq << 2;
  int hh = c / C;
  int s = src[e], d = dst[e];
  float w = eBuf[e * H + hh] / (sBuf[d * H + hh] + 1e-16f);
  v4f hv = *(const v4f*)(h + (size_t)s * H * C + c);
  float* op = out + (size_t)d * H * C + c;
  atomicAdd(op + 0, hv.x * w);
  atomicAdd(op + 1, hv.y * w);
  atomicAdd(op + 2, hv.z * w);
  atomicAdd(op + 3, hv.w * w);
}

// ------------------------------- ELU in place (float4) ---------------------
__global__ void elu_ip4(float* __restrict__ x, int nq) {
  int i = blockIdx.x * blockDim.x + threadIdx.x;
  if (i >= nq) return;
  v4f v = ((v4f*)x)[i];
  v.x = eluf(v.x); v.y = eluf(v.y); v.z = eluf(v.z); v.w = eluf(v.w);
  ((v4f*)x)[i] = v;
}

// ---------------- final: ELU + fc dot, one wave32 per node -----------------
__global__ void final_fc(const float* __restrict__ o2, const float* __restrict__ fcw,
                         const float* __restrict__ fcb, float* __restrict__ y,
                         int N, int C) {
  int wave = (blockIdx.x * blockDim.x + threadIdx.x) >> 5;
  int lane = threadIdx.x & 31;
  if (wave >= N) return;
  float acc = 0.f;
  for (int c = lane; c < C; c += 32) {
    acc += eluf(o2[(size_t)wave * C + c]) * fcw[c];
  }
  for (int off = 16; off > 0; off >>= 1) acc += __shfl_down(acc, off, 32);
  if (lane == 0) y[wave] = acc + fcb[0];
}

// ===========================================================================
extern "C" void kernel_launch(void* const* d_in, const int* in_sizes, int n_in,
                              void* d_out, int out_size, void* d_ws, size_t ws_size,
                              hipStream_t stream) {
  const float* x    = (const float*)d_in[0];   // [N,128]
  const int*   ei   = (const int*)d_in[1];     // [2,E]
  const float* W1   = (const float*)d_in[2];   // [128,256]
  const float* atl1 = (const float*)d_in[3];   // [4,64]
  const float* atr1 = (const float*)d_in[4];
  const float* W2   = (const float*)d_in[5];   // [256,64]
  const float* atl2 = (const float*)d_in[6];   // [64]
  const float* atr2 = (const float*)d_in[7];
  const float* fcw  = (const float*)d_in[8];   // [64]
  const float* fcb  = (const float*)d_in[9];   // [1]
  float* y = (float*)d_out;                    // [N]

  const int* src = ei;        // edge_index[0]
  const int* dst = ei + EE;   // edge_index[1]

  // ---- workspace layout (floats) ----
  float* w = (float*)d_ws;
  float* h1  = w; w += (size_t)NN * F1;     // GEMM1 out        [N,256]
  float* o1  = w; w += (size_t)NN * F1;     // L1 agg / ELU     [N,256]
  float* h2  = w; w += (size_t)NN * HID;    // GEMM2 out        [N,64]
  float* o2  = w; w += (size_t)NN * HID;    // L2 agg           [N,64]
  float* al1 = w; w += (size_t)NN * H1;
  float* ar1 = w; w += (size_t)NN * H1;
  float* al2 = w; w += NN;
  float* ar2 = w; w += NN;
  float* rw1 = w; w += NN;
  float* lg1 = w; w += NN;
  float* rw2 = w; w += NN;
  float* lg2 = w; w += NN;
  float* s1  = w; w += (size_t)NN * H1;
  float* s2  = w; w += NN;
  unsigned* m1 = (unsigned*)w; w += (size_t)NN * H1;
  unsigned* m2 = (unsigned*)w; w += NN;
  float* aE1 = w; w += (size_t)EE * H1;     // per-edge alpha   [E,4]
  float* aE2 = w; w += EE;                  // per-edge alpha   [E]
  float* sum1 = w; w += 1;
  float* sum2 = w; w += 1;

  const int TB = 256;
  auto blk = [](long long n, int t) { return (unsigned)((n + t - 1) / t); };

  // ================= Layer 1 (H=4, C=64, F=256, K=128) =====================
  fill_f32<<<blk((size_t)NN * F1, TB), TB, 0, stream>>>(o1, 0.f, NN * F1);
  fill_f32<<<blk(NN * H1, TB), TB, 0, stream>>>(s1, 0.f, NN * H1);
  fill_u32<<<blk(NN * H1, TB), TB, 0, stream>>>(m1, ORD_NEG_INF, NN * H1);

  rw_reduce<<<1, TB, 0, stream>>>(x, NN, DIN, rw1, sum1);
  log_rw<<<blk(NN, TB), TB, 0, stream>>>(rw1, sum1, lg1, NN);

  {   // h1 = x @ W1 : 3125 m-tiles * 4 col-quads = 12500 waves
    int waves = (NN / 16) * (F1 / 64);
    gemm_wmma_f32<<<blk((long long)waves * 32, TB), TB, 0, stream>>>(x, W1, h1, NN, DIN, F1);
  }
  node_attn<<<blk(NN * H1, TB), TB, 0, stream>>>(h1, atl1, atr1, al1, ar1, NN, H1, HID);

  edge_alpha<<<blk((long long)EE * H1, TB), TB, 0, stream>>>(src, dst, al1, ar1, lg1, aE1, m1, EE, H1);
  edge_exp<<<blk((long long)EE * H1, TB), TB, 0, stream>>>(dst, aE1, m1, s1, EE, H1);
  edge_aggregate4<<<blk((long long)EE * (F1 / 4), TB), TB, 0, stream>>>(src, dst, h1, aE1, s1, o1, EE, H1, HID);

  elu_ip4<<<blk((size_t)NN * F1 / 4, TB), TB, 0, stream>>>(o1, NN * F1 / 4);

  // ================= Layer 2 (H=1, C=64, F=64, K=256) ======================
  fill_f32<<<blk((size_t)NN * HID, TB), TB, 0, stream>>>(o2, 0.f, NN * HID);
  fill_f32<<<blk(NN, TB), TB, 0, stream>>>(s2, 0.f, NN);
  fill_u32<<<blk(NN, TB), TB, 0, stream>>>(m2, ORD_NEG_INF, NN);

  rw_reduce<<<1, TB, 0, stream>>>(o1, NN, F1, rw2, sum2);
  log_rw<<<blk(NN, TB), TB, 0, stream>>>(rw2, sum2, lg2, NN);

  {   // h2 = elu(o1) @ W2 : 3125 waves
    int waves = (NN / 16) * 1;
    gemm_wmma_f32<<<blk((long long)waves * 32, TB), TB, 0, stream>>>(o1, W2, h2, NN, F1, HID);
  }
  node_attn<<<blk(NN, TB), TB, 0, stream>>>(h2, atl2, atr2, al2, ar2, NN, 1, HID);

  edge_alpha<<<blk(EE, TB), TB, 0, stream>>>(src, dst, al2, ar2, lg2, aE2, m2, EE, 1);
  edge_exp<<<blk(EE, TB), TB, 0, stream>>>(dst, aE2, m2, s2, EE, 1);
  edge_aggregate4<<<blk((long long)EE * (HID / 4), TB), TB, 0, stream>>>(src, dst, h2, aE2, s2, o2, EE, 1, HID);

  // ================= head: ELU + fc (wave32 per node) ======================
  final_fc<<<blk((long long)NN * 32, TB), TB, 0, stream>>>(o2, fcw, fcb, y, NN, HID);
}